// ModelNew_50388556316945
// MI455X (gfx1250) — compile-verified
//
#include <hip/hip_runtime.h>
#include <hip/hip_bf16.h>
#include <math.h>

typedef float v2f __attribute__((ext_vector_type(2)));
typedef float v8f __attribute__((ext_vector_type(8)));

#define CIN    3
#define COUT   24
#define DIM    32
#define DOUT   30
#define KTOT   81          // CIN * 3*3*3
#define KSTEPS 21          // ceil(81 / 4)

#define SLAB_F   (CIN * DIM * 3 * DIM)   // [ci][z][hy][x] = 9216 floats
#define SLAB_PAD 16

__global__ __launch_bounds__(128)
void conv3d_min_softmax_wmma(const float* __restrict__ x,
                             const float* __restrict__ weight,
                             const float* __restrict__ bias,
                             float* __restrict__ out,
                             int N) {
    __shared__ float4 slab4[(SLAB_F + SLAB_PAD) / 4];
    __shared__ float  tile[32 * 32];
    __shared__ float  rmax[32];
    __shared__ float  rinv[32];
    float* slab = (float*)slab4;

    const int h    = blockIdx.x;      // output row 0..29
    const int n    = blockIdx.y;      // batch
    const int tid  = threadIdx.x;     // 0..127 (4 waves)
    const int lane = tid & 31;
    const int half = lane >> 4;       // lane-half selects K/M sub-rows per ISA layout
    const int lm   = lane & 15;
    const int wave = tid >> 5;
    const int mt   = wave & 1;        // M tile: w 0-15 / 16-31
    const int nt   = wave >> 1;       // N tile: c 0-15 / 16-31
    const int mbase = mt * 16;

    // Zero the slab pad so any (harmless) over-reads for padded rows are finite.
    if (tid < SLAB_PAD) slab[SLAB_F + tid] = 0.0f;

    // ---- Stage input slab x[n][ci][z][h+hy][0..31] into LDS via CDNA5 async
    //      global->LDS copies (ASYNCcnt-tracked, no VGPR round-trip). ----
    // Generic pointers to __shared__ carry the LDS byte offset in their low
    // 32 bits (LDS aperture lives in the high 32), so truncation yields the
    // per-lane LDS destination address for the async op.
    {
        const uint32_t lds_base = (uint32_t)(uintptr_t)slab;
        for (int v = tid; v < SLAB_F / 4; v += 128) {   // 2304 b128 copies, uniform trip
            int row = v >> 3;                 // 32-float rows: ci*96 + z*3 + hy
            int q   = v & 7;
            int ci  = row / 96;
            int rz  = row - ci * 96;
            int z   = rz / 3;
            int hy  = rz - z * 3;
            const float* g = x + (((size_t)n * CIN + ci) * DIM + z) * DIM * DIM
                               + (size_t)(h + hy) * DIM + (size_t)q * 4;
            uint32_t lds_dst = lds_base + (uint32_t)v * 16u;
            uint64_t gaddr   = (uint64_t)(uintptr_t)g;
            asm volatile("global_load_async_to_lds_b128 %0, %1, off"
                         :: "v"(lds_dst), "v"(gaddr)
                         : "memory");
        }
    }

    // ---- Precompute B fragments (weights) and A LDS offsets into registers
    //      (overlaps with the in-flight async copies above) ----
    // K index k -> (ci, kd, kh, kw); A addr = ci*3072 + (d+kd)*96 + kh*32 + (m+kw)
    float bfrag[KSTEPS][2];
    int   aoff [KSTEPS][2];
    const int cch = nt * 16 + lm;     // channel this lane owns in B/C/D (0..31, 24+ padded)
    const int m   = mbase + lm;       // w position this lane owns in A rows
#pragma unroll
    for (int s = 0; s < KSTEPS; ++s) {
#pragma unroll
        for (int j = 0; j < 2; ++j) {
            int k = 4 * s + j + 2 * half;
            if (k < KTOT) {
                int ci = k / 27;
                int r  = k - ci * 27;
                int kd = r / 9;
                int r2 = r - kd * 9;
                int kh = r2 / 3;
                int kw = r2 - kh * 3;
                aoff [s][j] = ci * 3072 + kd * 96 + kh * 32 + m + kw;
                bfrag[s][j] = (cch < COUT) ? weight[cch * KTOT + k] : 0.0f;
            } else {
                aoff [s][j] = m;      // safe in-bounds dummy; B=0 kills the product
                bfrag[s][j] = 0.0f;
            }
        }
    }

    // Drain this wave's async copies, then make all waves' LDS writes visible.
    asm volatile("s_wait_asynccnt 0x0" ::: "memory");
    __syncthreads();

    // ---- Conv GEMM per depth, running min over depth (full EXEC: no divergence) ----
    v8f vmin;
#pragma unroll
    for (int v = 0; v < 8; ++v) vmin[v] = INFINITY;

    for (int d = 0; d < DOUT; ++d) {
        v8f acc = {};
        const int dbase = d * 96;
#pragma unroll
        for (int s = 0; s < KSTEPS; ++s) {
            v2f a, b;
            a.x = slab[aoff[s][0] + dbase];
            a.y = slab[aoff[s][1] + dbase];
            b.x = bfrag[s][0];
            b.y = bfrag[s][1];
            // full-precision fp32 matrix op: v_wmma_f32_16x16x4_f32
            acc = __builtin_amdgcn_wmma_f32_16x16x4_f32(
                      false, a, false, b, (short)0, acc, false, false);
        }
#pragma unroll
        for (int v = 0; v < 8; ++v) vmin[v] = fminf(vmin[v], acc[v]);
    }

    // ---- Add bias, stage 32x24 (padded to 32) tile to LDS ----
    const float bi = (cch < COUT) ? bias[cch] : 0.0f;
#pragma unroll
    for (int v = 0; v < 8; ++v) {
        int mrow = mbase + v + 8 * half;   // C/D layout: VGPR v -> M = v + 8*lane_half
        tile[mrow * 32 + cch] = vmin[v] + bi;
    }
    __syncthreads();

    // ---- Softmax over 24 channels per valid w row ----
    if (tid < DOUT) {
        float mx = -INFINITY;
        for (int c = 0; c < COUT; ++c) mx = fmaxf(mx, tile[tid * 32 + c]);
        float sum = 0.0f;
        for (int c = 0; c < COUT; ++c) sum += __expf(tile[tid * 32 + c] - mx);
        rmax[tid] = mx;
        rinv[tid] = 1.0f / sum;
    }
    __syncthreads();

    for (int idx = tid; idx < DOUT * COUT; idx += 128) {
        int mrow = idx / COUT;
        int c    = idx - mrow * COUT;
        float val = __expf(tile[mrow * 32 + c] - rmax[mrow]) * rinv[mrow];
        out[(((size_t)n * COUT + c) * DOUT + h) * DOUT + mrow] = val;
    }
}

extern "C" void kernel_launch(void* const* d_in, const int* in_sizes, int n_in,
                              void* d_out, int out_size, void* d_ws, size_t ws_size,
                              hipStream_t stream) {
    const float* x      = (const float*)d_in[0];
    const float* weight = (const float*)d_in[1];
    const float* bias   = (const float*)d_in[2];
    float* out          = (float*)d_out;

    const int N = in_sizes[0] / (CIN * DIM * DIM * DIM);   // 256

    dim3 grid(DOUT, N);   // (h, n)
    dim3 block(128);      // 4 wave32
    conv3d_min_softmax_wmma<<<grid, block, 0, stream>>>(x, weight, bias, out, N);
}